// ClusterAssignment_28630251995897
// MI455X (gfx1250) — compile-verified
//
#include <hip/hip_runtime.h>
#include <hip/hip_bf16.h>
#include <math.h>

// ---------------------------------------------------------------------------
// MI455X (gfx1250) fused cluster-assignment + exact 1.5-entmax.
//
// Roofline: 192 MB HBM traffic (~8.2us @ 23.3TB/s) vs 4.3 GFLOP fp32 GEMM
// (V_WMMA_F32_16X16X4_F32, 2048 FLOP/inst) -> memory-bound when fused.
// Single pass: batch tile -> LDS via GLOBAL_LOAD_ASYNC_TO_LDS_B128 (ASYNCcnt,
// no VGPR bounce) -> WMMA -> square/normalize -> wave-cooperative Newton
// solve of the entmax tau (sum_i max(y_i - tau,0)^2 == 1, equivalent to the
// reference's sort-based construction) -> coalesced b64 stores.
// ---------------------------------------------------------------------------

typedef __attribute__((ext_vector_type(2))) float v2f;
typedef __attribute__((ext_vector_type(8))) float v8f;
typedef __attribute__((ext_vector_type(4))) int   v4i;

#define D_DIM       128   // feature dim
#define K_CL        64    // clusters
#define ROWS_PER_WG 128   // 8 waves * 16 rows
#define A_STRIDE    132   // LDS row stride (floats); 132 % 64 == 4 -> conflict-free
#define C_STRIDE    132
#define S_STRIDE    66    // score rows: 66 % 64 == 2, b64-aligned
#define NEWTON_IT   20

#if __has_builtin(__builtin_amdgcn_global_load_async_to_lds_b128)
#define HAVE_ASYNC_LDS 1
typedef __attribute__((address_space(1))) v4i* glb_b128_p;
typedef __attribute__((address_space(3))) v4i* lds_b128_p;
#endif

__device__ __forceinline__ void wait_async_lds() {
#if defined(HAVE_ASYNC_LDS)
#if __has_builtin(__builtin_amdgcn_s_wait_asynccnt)
  __builtin_amdgcn_s_wait_asynccnt(0);
#else
  asm volatile("s_wait_asynccnt 0" ::: "memory");
#endif
#endif
}

__device__ __forceinline__ float wave_max(float v) {
#pragma unroll
  for (int m = 16; m >= 1; m >>= 1) v = fmaxf(v, __shfl_xor(v, m, 32));
  return v;
}
__device__ __forceinline__ float wave_sum(float v) {
#pragma unroll
  for (int m = 16; m >= 1; m >>= 1) v += __shfl_xor(v, m, 32);
  return v;
}

// ---------------------------------------------------------------------------
// Main kernel: one WG = 128 rows x 64 clusters, fused GEMM + entmax15.
// ---------------------------------------------------------------------------
__global__ __launch_bounds__(256)
void fused_assign_entmax(const float* __restrict__ batch,
                         const float* __restrict__ centers,
                         float* __restrict__ out, int Mrows) {
  extern __shared__ float smem[];
  float* At    = smem;                          // 128 x 132 batch tile
  float* Ct    = At + ROWS_PER_WG * A_STRIDE;   // 64 x 132 centers tile
  float* norms = Ct + K_CL * C_STRIDE;          // 64 center norms

  const int tid = threadIdx.x;
  const int w   = tid >> 5;          // wave id 0..7
  const int L   = tid & 31;          // lane id
  const int m0  = blockIdx.x * ROWS_PER_WG;

  // ---- stage centers (64x128) and batch tile (128x128) into LDS ----------
  // Preferred path: GLOBAL_LOAD_ASYNC_TO_LDS_B128 (ASYNCcnt, no VGPR bounce).
  {
    const float4* csrc = (const float4*)centers;
    const float4* bsrc = (const float4*)(batch + (size_t)m0 * D_DIM);
#pragma unroll
    for (int i = 0; i < 8; ++i) {              // centers: 2048 float4 / 256 thr
      int g4 = tid + i * 256;
      int g  = g4 * 4;
      int row = g >> 7, d = g & 127;
#if defined(HAVE_ASYNC_LDS)
      __builtin_amdgcn_global_load_async_to_lds_b128(
          (glb_b128_p)(csrc + g4), (lds_b128_p)&Ct[row * C_STRIDE + d], 0, 0);
#else
      *(float4*)&Ct[row * C_STRIDE + d] = csrc[g4];
#endif
    }
#pragma unroll
    for (int i = 0; i < 16; ++i) {             // batch tile: 4096 float4 / 256 thr
      int g4 = tid + i * 256;
      int g  = g4 * 4;
      int row = g >> 7, d = g & 127;
      if (m0 + row < Mrows) {
#if defined(HAVE_ASYNC_LDS)
        __builtin_amdgcn_global_load_async_to_lds_b128(
            (glb_b128_p)(bsrc + g4), (lds_b128_p)&At[row * A_STRIDE + d], 0, 0);
#else
        *(float4*)&At[row * A_STRIDE + d] = bsrc[g4];
#endif
      } else {
        *(float4*)&At[row * A_STRIDE + d] = make_float4(0.f, 0.f, 0.f, 0.f);
      }
    }
  }
  wait_async_lds();
  __syncthreads();

  // ---- per-cluster norms ----
  if (tid < K_CL) {
    float s = 0.f;
#pragma unroll 4
    for (int d = 0; d < D_DIM; ++d) { float c = Ct[tid * C_STRIDE + d]; s += c * c; }
    norms[tid] = sqrtf(s);
  }
  __syncthreads();

  // ---- WMMA f32 16x16x4: wave w owns rows 16w..16w+15, 4 N-tiles of 16 ----
  // A frag (16x4 MxK):  lane L -> M = L%16, VGPR j -> K = 2*(L/16)+j
  // B frag (4x16 KxN):  lane L -> N = L%16, VGPR j -> K = 2*(L/16)+j
  // C tile: lane L, VGPR v -> N = L%16, M = v + 8*(L/16)
  v8f acc[4];
#pragma unroll
  for (int t = 0; t < 4; ++t) acc[t] = v8f{};

  const int nIdx  = L & 15;
  const int kHalf = 2 * (L >> 4);
  const int aBase = (16 * w + nIdx) * A_STRIDE;

#pragma unroll 4
  for (int kk = 0; kk < D_DIM / 4; ++kk) {
    const int dbase = 4 * kk + kHalf;
    v2f a = *(const v2f*)&At[aBase + dbase];
#pragma unroll
    for (int t = 0; t < 4; ++t) {
      v2f b = *(const v2f*)&Ct[(16 * t + nIdx) * C_STRIDE + dbase];
      acc[t] = __builtin_amdgcn_wmma_f32_16x16x4_f32(
          /*neg_a=*/false, a, /*neg_b=*/false, b,
          /*c_mod=*/(short)0, acc[t], /*reuse_a=*/false, /*reuse_b=*/false);
    }
  }

  // ---- square / normalize, transpose to row-major scores in wave-private
  //      LDS (alias onto this wave's own A-tile region: 16*132 >= 16*66) ----
  float* Sw = At + w * (16 * A_STRIDE);
  const int mHi = 8 * (L >> 4);
#pragma unroll
  for (int t = 0; t < 4; ++t) {
    float rn = 1.0f / norms[16 * t + nIdx];
#pragma unroll
    for (int v = 0; v < 8; ++v) {
      float c = acc[t][v];
      Sw[(mHi + v) * S_STRIDE + 16 * t + nIdx] = c * c * rn;
    }
  }

  // ---- wave-cooperative exact 1.5-entmax per row (K=64 -> 2 vals/lane) ----
  for (int r = 0; r < 16; ++r) {
    v2f s2 = *(const v2f*)&Sw[r * S_STRIDE + 2 * L];
    float mx = wave_max(fmaxf(s2.x, s2.y));
    float y0 = 0.5f * (s2.x - mx);
    float y1 = 0.5f * (s2.y - mx);
    float tau = -1.0f;                         // f(-1) >= 1, f(0) = 0
#pragma unroll
    for (int it = 0; it < NEWTON_IT; ++it) {
      float t0 = fmaxf(y0 - tau, 0.f), t1 = fmaxf(y1 - tau, 0.f);
      float f = wave_sum(t0 * t0 + t1 * t1);
      float g = wave_sum(t0 + t1);
      tau += (f - 1.0f) / (2.0f * g + 1e-30f);
    }
    float p0 = fmaxf(y0 - tau, 0.f); p0 *= p0;
    float p1 = fmaxf(y1 - tau, 0.f); p1 *= p1;
    int grow = m0 + 16 * w + r;
    if (grow < Mrows) {
      v2f p; p.x = p0; p.y = p1;
      *(v2f*)&out[(size_t)grow * K_CL + 2 * L] = p;   // coalesced b64 stores
    }
  }
}

// ---------------------------------------------------------------------------
// Tiny side kernel: lossp (|| (PP^T - I)^2 ||_F / K) and lossd (pairwise).
// One WG, 64 threads (one per cluster row).
// ---------------------------------------------------------------------------
__global__ __launch_bounds__(64)
void loss_kernel(const float* __restrict__ centers, float* __restrict__ scalars) {
  __shared__ float C[K_CL][130];   // centers, later overwritten with P rows
  __shared__ float Mm[K_CL][66];   // PP^T - I
  __shared__ float red[K_CL];

  const int k = threadIdx.x;
  for (int d = 0; d < D_DIM; ++d) C[k][d] = centers[k * D_DIM + d];
  __syncthreads();

  // lossd: sum over pairs j>i of ||c_i - c_j + 1e-6||
  float accd = 0.f;
  for (int j = k + 1; j < K_CL; ++j) {
    float s = 0.f;
    for (int d = 0; d < D_DIM; ++d) {
      float df = C[k][d] - C[j][d] + 1e-6f;
      s += df * df;
    }
    accd += sqrtf(s);
  }
  red[k] = accd;
  __syncthreads();
  if (k == 0) {
    float t = 0.f;
    for (int i = 0; i < K_CL; ++i) t += red[i];
    float total = 1e-10f + t;
    scalars[1] = 1.0f / total / (float)(K_CL * (K_CL - 1) / 2);
  }
  __syncthreads();

  // entmax15 of row k over D=128 (same Newton solve), overwrite in place
  {
    float mx = -INFINITY;
    for (int d = 0; d < D_DIM; ++d) mx = fmaxf(mx, C[k][d]);
    float tau = -1.0f;
    for (int it = 0; it < 24; ++it) {
      float f = 0.f, g = 0.f;
      for (int d = 0; d < D_DIM; ++d) {
        float t0 = fmaxf(0.5f * (C[k][d] - mx) - tau, 0.f);
        f += t0 * t0; g += t0;
      }
      tau += (f - 1.0f) / (2.0f * g + 1e-30f);
    }
    for (int d = 0; d < D_DIM; ++d) {
      float t0 = fmaxf(0.5f * (C[k][d] - mx) - tau, 0.f);
      C[k][d] = t0 * t0;
    }
  }
  __syncthreads();

  // M = P P^T - I
  for (int j = 0; j < K_CL; ++j) {
    float s = 0.f;
    for (int d = 0; d < D_DIM; ++d) s += C[k][d] * C[j][d];
    Mm[k][j] = s - (k == j ? 1.0f : 0.f);
  }
  __syncthreads();

  // || M @ M ||_F
  float accp = 0.f;
  for (int j = 0; j < K_CL; ++j) {
    float s = 0.f;
    for (int l = 0; l < K_CL; ++l) s += Mm[k][l] * Mm[l][j];
    accp += s * s;
  }
  red[k] = accp;
  __syncthreads();
  if (k == 0) {
    float t = 0.f;
    for (int i = 0; i < K_CL; ++i) t += red[i];
    scalars[0] = sqrtf(t) / (float)K_CL;
  }
}

// ---------------------------------------------------------------------------
extern "C" void kernel_launch(void* const* d_in, const int* in_sizes, int n_in,
                              void* d_out, int out_size, void* d_ws, size_t ws_size,
                              hipStream_t stream) {
  const float* batch   = (const float*)d_in[0];   // [B,N,D] f32
  const float* centers = (const float*)d_in[1];   // [K,D]   f32
  float* out = (float*)d_out;                     // [B,N,K] ++ [lossp, lossd]

  const int Mrows = in_sizes[0] / D_DIM;          // B*N rows
  const int nblk  = (Mrows + ROWS_PER_WG - 1) / ROWS_PER_WG;
  const size_t lds_bytes =
      (size_t)(ROWS_PER_WG * A_STRIDE + K_CL * C_STRIDE + K_CL) * sizeof(float);

  fused_assign_entmax<<<nblk, 256, lds_bytes, stream>>>(batch, centers, out, Mrows);
  loss_kernel<<<1, 64, 0, stream>>>(centers, out + (size_t)Mrows * K_CL);
}